// emitter_lightAccu_66907000537247
// MI455X (gfx1250) — compile-verified
//
#include <hip/hip_runtime.h>
#include <math.h>

#define IM_H 240
#define IM_W 320
#define ENV_R 120
#define ENV_C 160
#define NPIX (ENV_R*ENV_C)          // 19200 per batch
#define NG 384
#define BATCH 2

// dynamic-LDS partition (in floats)
#define ENV_PAD 388                 // 384 env words + 4 pad (16B aligned, rotates banks)
#define PHASE_G 96                  // grids processed per phase
#define ENV_WORDS (32*ENV_PAD)      // 12416
#define ACC_WORDS (3*PHASE_G*32)    // 9216
#define VEC_WORDS (PHASE_G*32*3)    // 9216
#define SMEM_WORDS (ENV_WORDS+ACC_WORDS+VEC_WORDS)  // 30848 -> 123,392 B

typedef float v2f __attribute__((ext_vector_type(2)));
typedef float v8f __attribute__((ext_vector_type(8)));
typedef int   v4i __attribute__((ext_vector_type(4)));

#if defined(__has_builtin)
#if __has_builtin(__builtin_amdgcn_global_load_async_to_lds_b128)
#define ASYNC_LDS 1
typedef __attribute__((address_space(1))) v4i* gbl_v4i_p;   // global src
typedef __attribute__((address_space(3))) v4i* lds_v4i_p;   // LDS dst
#endif
#endif

// ---------------------------------------------------------------- utility
__global__ void k_init(float* wsCount) {
  if (blockIdx.x == 0 && threadIdx.x < BATCH) wsCount[threadIdx.x] = 0.f;
}

// ---------------------------------------------------------------- points
__global__ void k_points(const float* __restrict__ depth, const float* __restrict__ camK,
                         float* __restrict__ outPoints) {
  const int NIM = IM_H * IM_W;
  const int t = blockIdx.x * blockDim.x + threadIdx.x;
  if (t >= BATCH * NIM) return;
  const int b = t / NIM;
  const int rem = t % NIM;
  const int v = rem / IM_W, u = rem % IM_W;
  const float fpix = camK[b*9 + 0] * (IM_W * 0.5f) / camK[b*9 + 2];
  const float z = -depth[t];
  const float x = -((float)u - IM_W * 0.5f) / fpix * z;
  const float y =  ((float)v - IM_H * 0.5f) / fpix * z;
  outPoints[((size_t)b*3 + 0)*NIM + rem] = x;
  outPoints[((size_t)b*3 + 1)*NIM + rem] = y;
  outPoints[((size_t)b*3 + 2)*NIM + rem] = z;
}

// -------------------------------------------------- per env-pixel precompute
// wsPix[pix*16]: [0..8] invA row-major, [9..11] pts_s, [12] mask
__global__ void k_pixel(const float* __restrict__ normalPred, const float* __restrict__ depth,
                        const float* __restrict__ camK, float* __restrict__ wsPix,
                        float* __restrict__ outMask, float* __restrict__ wsCount) {
  const int t = blockIdx.x * blockDim.x + threadIdx.x;
  if (t >= BATCH * NPIX) return;
  const int NIM = IM_H * IM_W;
  const int b = t / NPIX;
  const int rem = t % NPIX;
  const int r = rem / ENV_C, c0 = rem % ENV_C;
  const int v2 = 2*r, u2 = 2*c0;

  float n[3];
  #pragma unroll
  for (int ch = 0; ch < 3; ++ch) {
    const float* base = normalPred + ((size_t)b*3 + ch)*NIM + (size_t)v2*IM_W + u2;
    n[ch] = 0.25f * (base[0] + base[1] + base[IM_W] + base[IM_W + 1]);
  }
  const float nl = sqrtf(n[0]*n[0] + n[1]*n[1] + n[2]*n[2]) + 1e-6f;
  const float nx = n[0]/nl, ny = n[1]/nl, nz = n[2]/nl;
  // camy = normalize(up - (up.n) n), up=(0,1,0)
  float cy0 = -ny*nx, cy1 = 1.f - ny*ny, cy2 = -ny*nz;
  const float cl = sqrtf(cy0*cy0 + cy1*cy1 + cy2*cy2) + 1e-6f;
  cy0 /= cl; cy1 /= cl; cy2 /= cl;
  // camx = -normalize(cross(camy, normal))
  const float cr0 = cy1*nz - cy2*ny, cr1 = cy2*nx - cy0*nz, cr2 = cy0*ny - cy1*nx;
  const float crl = sqrtf(cr0*cr0 + cr1*cr1 + cr2*cr2) + 1e-6f;
  const float cx0 = -cr0/crl, cx1 = -cr1/crl, cx2 = -cr2/crl;
  // A columns = (camx, camy, normal) + 1e-6 I ; invert 3x3
  const float e = 1e-6f;
  const float a00 = cx0 + e, a01 = cy0,     a02 = nx;
  const float a10 = cx1,     a11 = cy1 + e, a12 = ny;
  const float a20 = cx2,     a21 = cy2,     a22 = nz + e;
  const float det = a00*(a11*a22 - a12*a21) - a01*(a10*a22 - a12*a20) + a02*(a10*a21 - a11*a20);
  const float id = 1.f / det;
  float* w = wsPix + (size_t)t * 16;
  w[0] = (a11*a22 - a12*a21)*id; w[1] = (a02*a21 - a01*a22)*id; w[2] = (a01*a12 - a02*a11)*id;
  w[3] = (a12*a20 - a10*a22)*id; w[4] = (a00*a22 - a02*a20)*id; w[5] = (a02*a10 - a00*a12)*id;
  w[6] = (a10*a21 - a11*a20)*id; w[7] = (a01*a20 - a00*a21)*id; w[8] = (a00*a11 - a01*a10)*id;
  // pts_s = points[:, :, ::2, ::2] at (v2,u2)
  const float fpix = camK[b*9 + 0] * (IM_W * 0.5f) / camK[b*9 + 2];
  const float z = -depth[(size_t)b*NIM + (size_t)v2*IM_W + u2];
  w[9]  = -((float)u2 - IM_W * 0.5f) / fpix * z;
  w[10] =  ((float)v2 - IM_H * 0.5f) / fpix * z;
  w[11] = z;
  const float mf = (z < -0.1f) ? 1.f : 0.f;
  w[12] = mf; w[13] = 0.f; w[14] = 0.f; w[15] = 0.f;
  outMask[t] = mf;
  atomicAdd(&wsCount[b], mf);   // integer-valued f32 adds: order-independent
}

// ------------------------------------------------------------ main fused pass
// Block = 32 waves; wave owns one env pixel; block owns 32 consecutive w pixels.
__global__ __launch_bounds__(1024)
void k_main(const float* __restrict__ env, const float* __restrict__ verts,
            const float* __restrict__ wsPix, float* __restrict__ wsPart,
            float* __restrict__ outAcc, float* __restrict__ outVec) {
  extern __shared__ float smem[];
  float* envS = smem;                      // 32 * ENV_PAD
  float* accS = smem + ENV_WORDS;          // 3 * PHASE_G * 32, [cg][w]
  float* vecS = accS + ACC_WORDS;          // PHASE_G * 32 * 3, [gl][w][c]

  const int tid  = threadIdx.x;
  const int wave = tid >> 5;
  const int lane = tid & 31;
  const int row  = lane & 15;              // WMMA A row / B column index
  const int hi   = lane >> 4;              // lane group (K half)

  const int P0  = blockIdx.x * 32;
  const int bB  = P0 / NPIX;
  const int remB = P0 % NPIX;
  const int hB  = remB / ENV_C;
  const int wB  = remB % ENV_C;            // ENV_C % 32 == 0: no row wrap
  const int P   = P0 + wave;

  // ---- stage this wave's envmap (3 x 128 f32, 1.5 KB) into LDS ----
  {
    float* dst = envS + wave * ENV_PAD;
    #pragma unroll
    for (int c = 0; c < 3; ++c) {
      const float* src = env + ((((size_t)bB*3 + c)*ENV_R + hB)*ENV_C + (wB + wave))*128 + lane*4;
#ifdef ASYNC_LDS
      __builtin_amdgcn_global_load_async_to_lds_b128(
          (gbl_v4i_p)src, (lds_v4i_p)(dst + c*128 + lane*4), 0, 0);
#else
      *(float4*)(dst + c*128 + lane*4) = *(const float4*)src;
#endif
    }
  }
#ifdef ASYNC_LDS
#if __has_builtin(__builtin_amdgcn_s_wait_asynccnt)
  __builtin_amdgcn_s_wait_asynccnt(0);
#else
  asm volatile("s_wait_asynccnt 0" ::: "memory");
#endif
#endif
  __syncthreads();

  // ---- per-pixel constants ----
  const float* wp = wsPix + (size_t)P * 16;
  const float psx = wp[9], psy = wp[10], psz = wp[11], maskf = wp[12];
  // WMMA A operand: 16x4 f32, A[row][k] = invA[row][k] (row<3,k<3) else 0.
  // Layout (ISA 7.12.2): lane L row=L%16; VGPR0 K = (L<16?0:2), VGPR1 K = (L<16?1:3).
  v2f Aop; Aop[0] = 0.f; Aop[1] = 0.f;
  if (row < 3) {
    Aop[0] = wp[row*3 + (hi ? 2 : 0)];
    if (!hi) Aop[1] = wp[row*3 + 1];     // K=3 column stays 0
  }
  const float* vb = verts + (size_t)bB * NG * 3;

  for (int ph = 0; ph < 4; ++ph) {
    for (int cgi = 0; cgi < PHASE_G/16; ++cgi) {
      const int gl = cgi*16 + row;       // local grid id within phase
      const int g  = ph*PHASE_G + gl;
      const float vx = vb[g*3 + 0] - psx;
      const float vy = vb[g*3 + 1] - psy;
      const float vz = vb[g*3 + 2] - psz;
      // B operand 4x16: column N = lane%16, K mirrors the A striping.
      v2f Bop; Bop[0] = hi ? vz : vx; Bop[1] = hi ? 0.f : vy;
      v8f czero = {};
      v8f d = __builtin_amdgcn_wmma_f32_16x16x4_f32(false, Aop, false, Bop,
                                                    (short)0, czero, false, false);
      if (!hi) {  // lanes 0..15 hold D rows M=0..2 in VGPR 0..2 for grid g
        const float lx = d[0], ly = d[1], lz = d[2];
        float* vs = vecS + gl*96 + wave*3;
        vs[0] = vx; vs[1] = vy; vs[2] = vz;
        const float il = 1.f / (sqrtf(lx*lx + ly*ly + lz*lz) + 1e-6f);
        const float theta = acosf(lz * il);
        const float phi = atan2f(ly, lx);           // == atan2(ly/sin,lx/sin), sin>0
        float az = (phi * 0.15915494309189535f + 0.5f) * 16.f;  // phi/2pi
        float el = theta * 5.0929581789406507f;                 // theta*16/pi
        az = fminf(fmaxf(az - 0.5f, 0.f), 15.f);
        el = fminf(fmaxf(el - 0.5f, 0.f), 7.f);
        const float x0 = floorf(az), y0 = floorf(el);
        const float fx = az - x0, fy = el - y0;
        const int x0i = (int)x0, y0i = (int)y0;
        const int x1i = min(x0i + 1, 15), y1i = min(y0i + 1, 7);
        const int i00 = y0i*16 + x0i, i10 = y0i*16 + x1i;
        const int i01 = y1i*16 + x0i, i11 = y1i*16 + x1i;
        const float w00 = (1.f-fx)*(1.f-fy), w10 = fx*(1.f-fy);
        const float w01 = (1.f-fx)*fy,       w11 = fx*fy;
        const float* e = envS + wave * ENV_PAD;
        #pragma unroll
        for (int c = 0; c < 3; ++c) {
          const float* ec = e + c*128;
          const float v = ec[i00]*w00 + ec[i10]*w10 + ec[i01]*w01 + ec[i11]*w11;
          accS[(c*PHASE_G + gl)*32 + wave] = v * maskf;
        }
      }
    }
    __syncthreads();

    // coalesced env_acc writeout + deterministic per-block mean partials
    for (int k = 0; k < ACC_WORDS/1024; ++k) {
      const int idx = k*1024 + tid;
      const int w  = idx & 31;            // == lane
      const int cg = idx >> 5;            // wave-uniform
      const int c  = cg / PHASE_G;
      const int g  = ph*PHASE_G + (cg % PHASE_G);
      const float v = accS[idx];
      outAcc[(((size_t)bB*3 + c)*NG + g)*NPIX + (size_t)hB*ENV_C + wB + w] = v;
      float s = v;
      #pragma unroll
      for (int m = 16; m >= 1; m >>= 1) s += __shfl_xor(s, m, 32);
      if (w == 0) wsPart[(size_t)blockIdx.x * (NG*3) + g*3 + c] = s;
    }
    // coalesced vec_to_t writeout: [g][h][w][c], (w,c) contiguous per g
    const size_t vbase0 = (size_t)bB * NG * NPIX;
    for (int k = 0; k < VEC_WORDS/1024; ++k) {
      const int idx = k*1024 + tid;
      const int gl  = idx / 96;
      const int r96 = idx % 96;           // = w*3 + c
      const int g   = ph*PHASE_G + gl;
      outVec[(vbase0 + (size_t)g*NPIX + (size_t)hB*ENV_C + wB)*3 + r96] = vecS[idx];
    }
    __syncthreads();
  }
}

// ------------------------------------------------------------ env_mean finish
__global__ void k_mean(const float* __restrict__ wsPart, const float* __restrict__ wsCount,
                       float* __restrict__ outMean) {
  const int t = blockIdx.x * blockDim.x + threadIdx.x;
  if (t >= BATCH * NG * 3) return;
  const int b = t / (NG*3);
  const int r = t % (NG*3);
  const int blocksPerB = NPIX / 32;       // 600
  const float* p = wsPart + (size_t)b * blocksPerB * (NG*3) + r;
  float s = 0.f;
  for (int i = 0; i < blocksPerB; ++i) s += p[(size_t)i * (NG*3)];
  outMean[t] = s / (wsCount[b] + 1e-6f);
}

// ---------------------------------------------------------------- launcher
extern "C" void kernel_launch(void* const* d_in, const int* in_sizes, int n_in,
                              void* d_out, int out_size, void* d_ws, size_t ws_size,
                              hipStream_t stream) {
  const float* normalPred = (const float*)d_in[0];
  const float* depthPred  = (const float*)d_in[1];
  const float* env        = (const float*)d_in[2];
  const float* camK       = (const float*)d_in[3];
  const float* verts      = (const float*)d_in[4];

  float* out = (float*)d_out;
  const size_t N_ACC  = (size_t)BATCH*3*NG*NPIX;    // 44,236,800
  const size_t N_MEAN = (size_t)BATCH*NG*3;         // 2,304
  const size_t N_PTS  = (size_t)BATCH*3*IM_H*IM_W;  // 460,800
  const size_t N_MASK = (size_t)BATCH*NPIX;         // 38,400
  float* outAcc    = out;
  float* outMean   = outAcc + N_ACC;
  float* outPoints = outMean + N_MEAN;
  float* outMask   = outPoints + N_PTS;
  float* outVec    = outMask + N_MASK;

  float* wsPix   = (float*)d_ws;                       // BATCH*NPIX*16 = 614,400
  float* wsCount = wsPix + (size_t)BATCH*NPIX*16;      // 2
  float* wsPart  = wsCount + 2;                        // 1200 * 1152

  k_init<<<1, 64, 0, stream>>>(wsCount);
  k_points<<<(BATCH*IM_H*IM_W)/256, 256, 0, stream>>>(depthPred, camK, outPoints);
  k_pixel<<<(BATCH*NPIX)/256, 256, 0, stream>>>(normalPred, depthPred, camK,
                                                wsPix, outMask, wsCount);
  const size_t smemBytes = SMEM_WORDS * sizeof(float); // 123,392 B
  (void)hipFuncSetAttribute(reinterpret_cast<const void*>(k_main),
                            hipFuncAttributeMaxDynamicSharedMemorySize, (int)smemBytes);
  k_main<<<(BATCH*NPIX)/32, 1024, smemBytes, stream>>>(env, verts, wsPix, wsPart,
                                                       outAcc, outVec);
  k_mean<<<(BATCH*NG*3 + 255)/256, 256, 0, stream>>>(wsPart, wsCount, outMean);
}